// HPNLayer_23819888624045
// MI455X (gfx1250) — compile-verified
//
#include <hip/hip_runtime.h>
#include <hip/hip_bf16.h>
#include <math.h>

typedef __attribute__((ext_vector_type(2))) float v2f;
typedef __attribute__((ext_vector_type(8))) float v8f;

#define FEAT    64
#define HIDDEN  128
#define NPATHS  3
#define KLAYER  3
#define ALPHA   0.1f

// ---------------- elementwise / setup kernels ----------------

__global__ void fill_f32(float* __restrict__ p, float v, long n) {
  long i = (long)blockIdx.x * blockDim.x + threadIdx.x;
  if (i < n) p[i] = v;
}

__global__ void relu_k(const float* __restrict__ h, float* __restrict__ x, long n) {
  long i = (long)blockIdx.x * blockDim.x + threadIdx.x;
  if (i < n) x[i] = fmaxf(h[i], 0.0f);
}

// degrees for all paths at once: degO/degI are [NPATHS*N]
__global__ void degree_k(const long long* __restrict__ src,
                         const long long* __restrict__ dst,
                         float* __restrict__ degO, float* __restrict__ degI,
                         long E, int N) {
  long i = (long)blockIdx.x * blockDim.x + threadIdx.x;
  if (i >= (long)NPATHS * E) return;
  int  p = (int)(i / E);
  long e = i - (long)p * E;
  int s = (int)src[(long)p * E + e];
  int d = (int)dst[(long)p * E + e];
  atomicAdd(&degO[(long)p * N + s], 1.0f);
  atomicAdd(&degI[(long)p * N + d], 1.0f);
}

__global__ void norm_k(const float* __restrict__ deg, float* __restrict__ nrm, long n) {
  long i = (long)blockIdx.x * blockDim.x + threadIdx.x;
  if (i < n) nrm[i] = rsqrtf(fmaxf(deg[i], 1.0f));
}

// scaled[n,f] = feat[n,f] * norm_out[n]   (first propagation step only)
__global__ void scale_k(const float* __restrict__ feat, const float* __restrict__ nrm,
                        float* __restrict__ outp, long nf) {
  long i = (long)blockIdx.x * blockDim.x + threadIdx.x;
  if (i >= nf) return;
  outp[i] = feat[i] * nrm[i >> 6];   // FEAT == 64
}

// edge scatter-add: 16 lanes per edge, one float4 per lane (coalesced 256B/edge)
__global__ void scatter_k(const float* __restrict__ scaled, float* __restrict__ agg,
                          const long long* __restrict__ src,
                          const long long* __restrict__ dst, long E) {
  long t = (long)blockIdx.x * blockDim.x + threadIdx.x;
  if (t >= E * 16) return;
  long e = t >> 4;
  int  c = (int)(t & 15) << 2;
  int  s = (int)src[e];
  int  d = (int)dst[e];
  const float4 v = *(const float4*)(scaled + (long)s * FEAT + c);
  float* ap = agg + (long)d * FEAT + c;
  atomicAdd(ap + 0, v.x);
  atomicAdd(ap + 1, v.y);
  atomicAdd(ap + 2, v.z);
  atomicAdd(ap + 3, v.w);
}

// feat' = (1-a)*agg*norm_in[n] + a*x0 ; optionally also emit feat'*norm_out[n]
// (fuses next step's prescale into this pass, saving a full nf read+write)
__global__ void update_k(float* __restrict__ outf, float* __restrict__ outScaled,
                         const float* __restrict__ agg,
                         const float* __restrict__ x0,
                         const float* __restrict__ nrmI,
                         const float* __restrict__ nrmO,
                         int writeScaled, long nf) {
  long i = (long)blockIdx.x * blockDim.x + threadIdx.x;
  if (i >= nf) return;
  long node = i >> 6;
  float v = (1.0f - ALPHA) * agg[i] * nrmI[node] + ALPHA * x0[i];
  outf[i] = v;
  if (writeScaled) outScaled[i] = v * nrmO[node];
}

// ---------------- WMMA semantic-fusion GEMM ----------------
// One wave per 16-row tile of z (layout [P][N][FEAT]).
// hidden(16x128) = zTile(16x64) @ W1(64x128)   via V_WMMA_F32_16X16X4_F32
// s(16) = tanh(hidden + b1) @ W2 ; atomicAdd per-path score sum.
__global__ void fuse_gemm_k(const float* __restrict__ z,
                            const float* __restrict__ W1,
                            const float* __restrict__ b1,
                            const float* __restrict__ W2,
                            float* __restrict__ w_sum,
                            int N, int tilesPerPath) {
  int wave = (int)(((long)blockIdx.x * blockDim.x + threadIdx.x) >> 5);
  int lane = threadIdx.x & 31;
  int totalTiles = NPATHS * tilesPerPath;
  if (wave >= totalTiles) return;

  int p = wave / tilesPerPath;
  int t = wave - p * tilesPerPath;
  int rowBase = t * 16;

  int m    = lane & 15;   // row-in-tile for A frag / column-in-subtile for B,C,D
  int half = lane >> 4;   // K sub-select for A/B frags; +8 rows for C/D

  // Clamp instead of predicate: tail rows duplicate row N-1; each output row
  // depends only on its own A row, and tail rows are excluded from the score
  // accumulation below — so no EXEC divergence, single b64 A-loads.
  int row  = rowBase + m;
  int rowC = row < N ? row : (N - 1);
  const float* zrow = z + ((long)p * N + rowC) * FEAT;

  v8f acc[8];
#pragma unroll
  for (int j = 0; j < 8; ++j) acc[j] = (v8f){0.f,0.f,0.f,0.f,0.f,0.f,0.f,0.f};

  for (int k0 = 0; k0 < FEAT; k0 += 4) {
    int kk = k0 + 2 * half;
    const float2 a2 = *(const float2*)(zrow + kk);   // one global_load_b64
    v2f a; a.x = a2.x; a.y = a2.y;
#pragma unroll
    for (int j = 0; j < 8; ++j) {
      int col = j * 16 + m;
      v2f b;
      b.x = W1[(long)kk * HIDDEN + col];
      b.y = W1[(long)(kk + 1) * HIDDEN + col];
      acc[j] = __builtin_amdgcn_wmma_f32_16x16x4_f32(
          false, a, false, b, (short)0, acc[j], false, false);
    }
  }

  // Per-lane: 8 columns (j*16+m), 8 rows (r + 8*half)
  float part[8];
#pragma unroll
  for (int r = 0; r < 8; ++r) part[r] = 0.0f;
#pragma unroll
  for (int j = 0; j < 8; ++j) {
    int col = j * 16 + m;
    float bb = b1[col];
    float w2 = W2[col];
#pragma unroll
    for (int r = 0; r < 8; ++r)
      part[r] += tanhf(acc[j][r] + bb) * w2;
  }
  // reduce over 16 lanes within each half-wave
#pragma unroll
  for (int off = 8; off >= 1; off >>= 1) {
#pragma unroll
    for (int r = 0; r < 8; ++r)
      part[r] += __shfl_xor(part[r], off, 16);
  }
  if (m == 0) {
    float local = 0.0f;
#pragma unroll
    for (int r = 0; r < 8; ++r) {
      int M = r + 8 * half;
      if (rowBase + M < N) local += part[r];
    }
    atomicAdd(&w_sum[p], local);
  }
}

__global__ void softmax_k(const float* __restrict__ w_sum, float* __restrict__ beta, int N) {
  if (threadIdx.x != 0 || blockIdx.x != 0) return;
  float invN = 1.0f / (float)N;
  float w0 = w_sum[0] * invN, w1 = w_sum[1] * invN, w2 = w_sum[2] * invN;
  float mx = fmaxf(w0, fmaxf(w1, w2));
  float e0 = expf(w0 - mx), e1 = expf(w1 - mx), e2 = expf(w2 - mx);
  float s = e0 + e1 + e2;
  beta[0] = e0 / s; beta[1] = e1 / s; beta[2] = e2 / s;
}

__global__ void combine_k(const float* __restrict__ z, const float* __restrict__ beta,
                          float* __restrict__ outp, long nf) {
  long i = (long)blockIdx.x * blockDim.x + threadIdx.x;
  if (i >= nf) return;
  outp[i] = beta[0] * z[i] + beta[1] * z[i + nf] + beta[2] * z[i + 2 * nf];
}

// ---------------- host side ----------------

static inline unsigned gblocks(long n, int t) { return (unsigned)((n + t - 1) / t); }

extern "C" void kernel_launch(void* const* d_in, const int* in_sizes, int n_in,
                              void* d_out, int out_size, void* d_ws, size_t ws_size,
                              hipStream_t stream) {
  const float*      h   = (const float*)d_in[0];
  const long long*  src = (const long long*)d_in[1];   // int64 in reference
  const long long*  dst = (const long long*)d_in[2];
  const float*      W1  = (const float*)d_in[3];
  const float*      b1  = (const float*)d_in[4];
  const float*      W2  = (const float*)d_in[5];
  float*            out = (float*)d_out;

  const int  N  = in_sizes[0] / FEAT;
  const long E  = (long)in_sizes[1] / NPATHS;
  const long nf = (long)N * FEAT;

  // carve workspace
  char* ws = (char*)d_ws;
  float* x      = (float*)ws; ws += nf * sizeof(float);
  float* z      = (float*)ws; ws += (long)NPATHS * nf * sizeof(float);
  float* scaled = (float*)ws; ws += nf * sizeof(float);
  float* agg    = (float*)ws; ws += nf * sizeof(float);
  float* fbuf   = (float*)ws; ws += nf * sizeof(float);
  float* degO   = (float*)ws; ws += (long)NPATHS * N * sizeof(float);
  float* degI   = (float*)ws; ws += (long)NPATHS * N * sizeof(float);
  float* nrmO   = (float*)ws; ws += (long)NPATHS * N * sizeof(float);
  float* nrmI   = (float*)ws; ws += (long)NPATHS * N * sizeof(float);
  float* w_sum  = (float*)ws; ws += 4 * sizeof(float);
  float* beta   = (float*)ws; ws += 4 * sizeof(float);

  const int T = 256;

  // x = relu(h)
  relu_k<<<gblocks(nf, T), T, 0, stream>>>(h, x, nf);

  // degrees + norms (all paths)
  fill_f32<<<gblocks((long)NPATHS * N, T), T, 0, stream>>>(degO, 0.0f, (long)NPATHS * N);
  fill_f32<<<gblocks((long)NPATHS * N, T), T, 0, stream>>>(degI, 0.0f, (long)NPATHS * N);
  degree_k<<<gblocks((long)NPATHS * E, T), T, 0, stream>>>(src, dst, degO, degI, E, N);
  norm_k<<<gblocks((long)NPATHS * N, T), T, 0, stream>>>(degO, nrmO, (long)NPATHS * N);
  norm_k<<<gblocks((long)NPATHS * N, T), T, 0, stream>>>(degI, nrmI, (long)NPATHS * N);

  // APPNP propagation per path (prescale fused into update for steps >= 1)
  for (int p = 0; p < NPATHS; ++p) {
    const float* nO = nrmO + (long)p * N;
    const float* nI = nrmI + (long)p * N;
    // step 0 prescale of x
    scale_k<<<gblocks(nf, T), T, 0, stream>>>(x, nO, scaled, nf);
    for (int k = 0; k < KLAYER; ++k) {
      fill_f32<<<gblocks(nf, T), T, 0, stream>>>(agg, 0.0f, nf);
      scatter_k<<<gblocks(E * 16, T), T, 0, stream>>>(scaled, agg,
                                                      src + (long)p * E,
                                                      dst + (long)p * E, E);
      int   last = (k == KLAYER - 1);
      float* tgt = last ? (z + (long)p * nf) : fbuf;
      update_k<<<gblocks(nf, T), T, 0, stream>>>(tgt, scaled, agg, x, nI, nO,
                                                 /*writeScaled=*/!last, nf);
    }
  }

  // semantic fusion: WMMA GEMM -> per-path mean score -> softmax -> combine
  fill_f32<<<1, 32, 0, stream>>>(w_sum, 0.0f, 4);
  int tilesPerPath = (N + 15) / 16;
  long totalWaves  = (long)NPATHS * tilesPerPath;
  long gemmThreads = totalWaves * 32;
  fuse_gemm_k<<<gblocks(gemmThreads, T), T, 0, stream>>>(z, W1, b1, W2, w_sum, N, tilesPerPath);
  softmax_k<<<1, 32, 0, stream>>>(w_sum, beta, N);
  combine_k<<<gblocks(nf, T), T, 0, stream>>>(z, beta, out, nf);
}